// FocalDiceLoss_77653008712127
// MI455X (gfx1250) — compile-verified
//
#include <hip/hip_runtime.h>

#define NC    19
#define BATCH 8
#define HW    (512 * 512)
#define NPIX  (BATCH * HW)
#define PCOLS 48   // 19 inter + 19 denom + 3 scalars, padded to 3x16 for WMMA

typedef float v2f __attribute__((ext_vector_type(2)));
typedef float v8f __attribute__((ext_vector_type(8)));

// ---------------- wave32 sum reduction ----------------
#if __has_builtin(__builtin_amdgcn_update_dpp) && __has_builtin(__builtin_amdgcn_permlanex16)
// VALU-only reduction: DPP adds + cross-half permlanex16 (keeps the DS pipe free)
template <int CTRL>
__device__ __forceinline__ float dpp_add_step(float v) {
    int x = __float_as_int(v);
    int t = __builtin_amdgcn_update_dpp(0, x, CTRL, 0xf, 0xf, true);
    return v + __int_as_float(t);
}
__device__ __forceinline__ float wave_sum(float v) {
    v = dpp_add_step<0xB1>(v);   // quad_perm [1,0,3,2]  : xor 1
    v = dpp_add_step<0x4E>(v);   // quad_perm [2,3,0,1]  : xor 2
    v = dpp_add_step<0x141>(v);  // row_half_mirror      : xor 4
    v = dpp_add_step<0x140>(v);  // row_mirror           : xor 8
    int x = __float_as_int(v);   // xor 16: swap lanes 0-15 <-> 16-31
    int t = __builtin_amdgcn_permlanex16(x, x, 0x76543210, 0xfedcba98, false, false);
    return v + __int_as_float(t);
}
#else
__device__ __forceinline__ float wave_sum(float v) {
    v += __shfl_xor(v, 1, 32);
    v += __shfl_xor(v, 2, 32);
    v += __shfl_xor(v, 4, 32);
    v += __shfl_xor(v, 8, 32);
    v += __shfl_xor(v, 16, 32);
    return v;
}
#endif

// Kernel 1: streaming pass over logits; per-block partial sums of
// [inter[19] | denom[19] | ce_num | ce_den | focal_sum | pad...] -> part[block][48]
__global__ void __launch_bounds__(256) loss_partial(
    const float* __restrict__ logits, const int* __restrict__ target,
    const float* __restrict__ cw, const float* __restrict__ fa,
    float* __restrict__ part, int nblocks)
{
    __shared__ float sdata[8][PCOLS];
    __shared__ float cw_s[NC], fa_s[NC];
    const int tid = threadIdx.x;

    if (tid < NC) { cw_s[tid] = cw[tid]; fa_s[tid] = fa[tid]; }
    for (int i = tid; i < 8 * PCOLS; i += 256) ((float*)sdata)[i] = 0.f;
    __syncthreads();

    float inter[NC], denom[NC];
#pragma unroll
    for (int c = 0; c < NC; ++c) { inter[c] = 0.f; denom[c] = 0.f; }
    float ce_num = 0.f, ce_den = 0.f, fsum = 0.f;

    const int stride = nblocks * 256;
    for (int pix = blockIdx.x * 256 + tid; pix < NPIX; pix += stride) {
        const int b  = pix >> 18;            // pix / HW
        const int hw = pix & (HW - 1);
        const float* lp = logits + (size_t)b * NC * HW + hw;

        float v[NC];
#pragma unroll
        for (int c = 0; c < NC; ++c)         // coalesced, single-use -> NT hint
            v[c] = __builtin_nontemporal_load(lp + c * HW);

        float m = v[0];
#pragma unroll
        for (int c = 1; c < NC; ++c) m = fmaxf(m, v[c]);

        const int t = __builtin_nontemporal_load(target + pix);
        float lt = 0.f, et = 0.f, s = 0.f;
#pragma unroll
        for (int c = 0; c < NC; ++c) {
            if (c == t) lt = v[c];
            v[c] = __expf(v[c] - m);
            if (c == t) et = v[c];
            s += v[c];
        }
        const float inv_s  = 1.0f / s;
        const float log_pt = (lt - m) - __logf(s);
        const float pt     = et * inv_s;

        const float wt = cw_s[t], at = fa_s[t];
        ce_num += wt * (-log_pt);
        ce_den += wt;
        const float ptc = fmaxf(pt, 1e-8f);
        const float om  = 1.f - ptc;
        fsum += (-at) * om * om * log_pt;

#pragma unroll
        for (int c = 0; c < NC; ++c) {
            const float pc = v[c] * inv_s;
            denom[c] += (c == t) ? (pc + 1.f) : pc;   // prob_sums + counts
            inter[c] += (c == t) ? pc : 0.f;          // scatter-add of p_t
        }
    }

    // 32-lane reduction of each accumulator (VALU DPP ladder)
    const int lane = tid & 31, wid = tid >> 5;
#pragma unroll
    for (int c = 0; c < NC; ++c) {
        inter[c] = wave_sum(inter[c]);
        denom[c] = wave_sum(denom[c]);
    }
    ce_num = wave_sum(ce_num);
    ce_den = wave_sum(ce_den);
    fsum   = wave_sum(fsum);

    if (lane == 0) {
#pragma unroll
        for (int c = 0; c < NC; ++c) { sdata[wid][c] = inter[c]; sdata[wid][NC + c] = denom[c]; }
        sdata[wid][38] = ce_num; sdata[wid][39] = ce_den; sdata[wid][40] = fsum;
    }
    __syncthreads();

    // Cross-wave reduction via V_WMMA_F32_16X16X4_F32 with B = ones:
    // D[m][n] = sum_k A[m][k];  A[M][K] = sdata[wave K][col M]  (documented A layout)
    if (tid < 32) {
        const int m16 = tid & 15;
        const int hi  = tid >> 4;          // 0: K=0,1   1: K=2,3
        v2f bones; bones.x = 1.f; bones.y = 1.f;
        float* out = part + (size_t)blockIdx.x * PCOLS;
#pragma unroll
        for (int g = 0; g < 3; ++g) {
            const int col = g * 16 + m16;
            v2f a0, a1;
            a0.x = sdata[hi ? 2 : 0][col];  // first WMMA: waves 0..3
            a0.y = sdata[hi ? 3 : 1][col];
            a1.x = sdata[hi ? 6 : 4][col];  // second WMMA: waves 4..7
            a1.y = sdata[hi ? 7 : 5][col];
            v8f acc = {};
            acc = __builtin_amdgcn_wmma_f32_16x16x4_f32(false, a0, false, bones,
                                                        (short)0, acc, false, false);
            acc = __builtin_amdgcn_wmma_f32_16x16x4_f32(false, a1, false, bones,
                                                        (short)0, acc, false, false);
            // D layout: VGPR r = row M (lanes 0-15: M=r, lanes 16-31: M=r+8), replicated over N
            if (m16 == 0) {
                const int mbase = hi * 8;
#pragma unroll
                for (int r = 0; r < 8; ++r) out[g * 16 + mbase + r] = acc[r];
            }
        }
    }
}

// Kernel 2: reduce part[rows][48] column-wise -> colsum[48]
__global__ void __launch_bounds__(256) col_reduce(
    const float* __restrict__ part, float* __restrict__ colsum, int rows)
{
    const int col = blockIdx.x;
    float s = 0.f;
    for (int r = threadIdx.x; r < rows; r += 256) s += part[(size_t)r * PCOLS + col];
    s = wave_sum(s);
    __shared__ float w8[8];
    const int lane = threadIdx.x & 31, wid = threadIdx.x >> 5;
    if (lane == 0) w8[wid] = s;
    __syncthreads();
    if (threadIdx.x == 0) {
        float t = 0.f;
#pragma unroll
        for (int i = 0; i < 8; ++i) t += w8[i];
        colsum[col] = t;
    }
}

// Kernel 3: combine into final scalar loss
__global__ void finalize_loss(const float* __restrict__ colsum,
                              const float* __restrict__ cw,
                              float* __restrict__ out)
{
    if (threadIdx.x == 0 && blockIdx.x == 0) {
        const float ce    = colsum[38] / colsum[39];
        const float focal = colsum[40] * (1.0f / (float)NPIX);
        float sw = 0.f;
        for (int c = 0; c < NC; ++c) sw += cw[c];
        const float inv = 1.f / fmaxf(sw, 1e-8f);
        float ds = 0.f;
        for (int c = 0; c < NC; ++c)
            ds += (cw[c] * inv) * (2.f * colsum[c] + 1.f) / (colsum[NC + c] + 1.f);
        out[0] = 0.4f * ce + 0.3f * focal + 0.3f * (1.f - ds);
    }
}

extern "C" void kernel_launch(void* const* d_in, const int* in_sizes, int n_in,
                              void* d_out, int out_size, void* d_ws, size_t ws_size,
                              hipStream_t stream)
{
    const float* logits = (const float*)d_in[0];
    const int*   target = (const int*)d_in[1];
    const float* cw     = (const float*)d_in[2];
    const float* fa     = (const float*)d_in[3];
    float* ws = (float*)d_ws;

    long cap = (long)(ws_size / sizeof(float)) - PCOLS;
    int G = (int)(cap / PCOLS);
    if (G > 1024) G = 1024;   // 1024 blocks x 8 waves = 8192 waves; 8 pixels/thread
    if (G < 1)    G = 1;

    float* part   = ws;
    float* colsum = ws + (size_t)G * PCOLS;

    loss_partial<<<G, 256, 0, stream>>>(logits, target, cw, fa, part, G);
    col_reduce<<<PCOLS, 256, 0, stream>>>(part, colsum, G);
    finalize_loss<<<1, 64, 0, stream>>>(colsum, cw, (float*)d_out);
}